// PerformanceModel_69097433858797
// MI455X (gfx1250) — compile-verified
//
#include <hip/hip_runtime.h>
#include <hip/hip_bf16.h>

// PerformanceModel: out[i] = pw1(bins[idx[i,0]]) * pw2(bins[idx[i,1]])
// Key insight: pw1/pw2 have UNIFORM scalar bounds -> precompute them per bin
// into two 1024-entry LDS tables once per block. Inner loop per output is then
// 2 ds_load gathers + 1 v_mul. Memory-bound: 128MB idx read + 64MB out write
// -> ~8.2us floor @ 23.3 TB/s; issue time ~1.7us, so firmly HBM-bound.
// b128 non-temporal index loads, b128 non-temporal stores, wave32 blocks.

#define N_BINS 1024
#define EPSF   1e-4f

constexpr int TPB = 256;   // 8 wave32s per block
constexpr int OPT = 32;    // outputs per thread -> 8192 outputs per block

// clang vector types (required by __builtin_nontemporal_*)
typedef int   v4i __attribute__((ext_vector_type(4)));
typedef float v4f __attribute__((ext_vector_type(4)));

__device__ __forceinline__ float fast_sigmoid(float x) {
    // v_exp_f32 + v_rcp_f32 (~1 ulp), avoids the IEEE div expansion
    return __builtin_amdgcn_rcpf(1.0f + __expf(-x));
}

// piecewise ramp: t<=lo -> 1 ; t>hi -> 0 ; else (hi-t)*inv
// inv = 1/(hi-lo+eps) precomputed, c = hi*inv so the ramp is a single FMA.
__device__ __forceinline__ float pwf(float t, float lo, float hi,
                                     float inv, float c) {
    float mid = fmaf(-t, inv, c);          // (hi - t) * inv
    float r = (t <= lo) ? 1.0f : mid;
    return (t > hi) ? 0.0f : r;
}

__global__ __launch_bounds__(TPB) void PerformanceModel_kernel(
    const float* __restrict__ bin_centers,
    const float* __restrict__ lb1, const float* __restrict__ ub1,
    const float* __restrict__ lb2, const float* __restrict__ ub2,
    const int*   __restrict__ obs_idx,   // [N,2] interleaved
    float*       __restrict__ out,
    int n)                               // n = N_DIFFS
{
    __shared__ float s_p1[N_BINS];   // pw1 evaluated at every bin center
    __shared__ float s_p2[N_BINS];   // pw2 evaluated at every bin center

    // ---- Preamble: uniform scalar params (fast math), then build the two
    //      precomputed ramp tables: 256 threads x 4 bins each. ----
    const float a1 = lb1[0], b1 = ub1[0], a2 = lb2[0], b2 = ub2[0];
    const float l1 = fminf(a1, b1), u1 = fmaxf(a1, b1);
    const float l2 = fminf(a2, b2), u2 = fmaxf(a2, b2);
    const float nl1 = fast_sigmoid(l1), nu1 = fast_sigmoid(u1);
    const float nl2 = fast_sigmoid(l2), nu2 = fast_sigmoid(u2);
    const float inv1 = __builtin_amdgcn_rcpf(nu1 - nl1 + EPSF);
    const float inv2 = __builtin_amdgcn_rcpf(nu2 - nl2 + EPSF);
    const float c1 = nu1 * inv1;
    const float c2 = nu2 * inv2;

    {
        v4f b = ((const v4f*)bin_centers)[threadIdx.x];   // 4 bins/thread
        v4f p1, p2;
        p1.x = pwf(b.x, nl1, nu1, inv1, c1);
        p1.y = pwf(b.y, nl1, nu1, inv1, c1);
        p1.z = pwf(b.z, nl1, nu1, inv1, c1);
        p1.w = pwf(b.w, nl1, nu1, inv1, c1);
        p2.x = pwf(b.x, nl2, nu2, inv2, c2);
        p2.y = pwf(b.y, nl2, nu2, inv2, c2);
        p2.z = pwf(b.z, nl2, nu2, inv2, c2);
        p2.w = pwf(b.w, nl2, nu2, inv2, c2);
        ((v4f*)s_p1)[threadIdx.x] = p1;
        ((v4f*)s_p2)[threadIdx.x] = p2;
    }
    __syncthreads();

    // ---- Streaming phase: per output = 2 LDS gathers + 1 mul. ----
    const v4i* __restrict__ idx4 = (const v4i*)obs_idx;  // 2 index pairs each
    v4f*       __restrict__ out4 = (v4f*)out;            // 4 outputs each

    const int n4  = n >> 2;                     // number of v4f output slots
    const int spb = TPB * (OPT / 4);            // slots per block (2048)
    const int s0  = blockIdx.x * spb + threadIdx.x;

    if ((blockIdx.x + 1) * spb <= n4) {
        // Full block: no per-slot bounds checks (uniform scalar branch).
#pragma unroll
        for (int k = 0; k < OPT / 4; ++k) {
            const int s = s0 + k * TPB;
            v4i pa = __builtin_nontemporal_load(&idx4[2 * s]);
            v4i pb = __builtin_nontemporal_load(&idx4[2 * s + 1]);

            v4f r;
            r.x = s_p1[pa.x] * s_p2[pa.y];
            r.y = s_p1[pa.z] * s_p2[pa.w];
            r.z = s_p1[pb.x] * s_p2[pb.y];
            r.w = s_p1[pb.z] * s_p2[pb.w];

            __builtin_nontemporal_store(r, &out4[s]);
        }
    } else {
        // Partial block: per-slot checks.
#pragma unroll
        for (int k = 0; k < OPT / 4; ++k) {
            const int s = s0 + k * TPB;
            if (s < n4) {
                v4i pa = __builtin_nontemporal_load(&idx4[2 * s]);
                v4i pb = __builtin_nontemporal_load(&idx4[2 * s + 1]);

                v4f r;
                r.x = s_p1[pa.x] * s_p2[pa.y];
                r.y = s_p1[pa.z] * s_p2[pa.w];
                r.z = s_p1[pb.x] * s_p2[pb.y];
                r.w = s_p1[pb.z] * s_p2[pb.w];

                __builtin_nontemporal_store(r, &out4[s]);
            }
        }
    }

    // Sub-4 remainder (n % 4 elements), handled scalar by the last block.
    if (blockIdx.x == gridDim.x - 1) {
        const int rem = n & 3;
        if ((int)threadIdx.x < rem) {
            const int i  = (n & ~3) + threadIdx.x;
            const int i0 = obs_idx[2 * i];
            const int i1 = obs_idx[2 * i + 1];
            out[i] = s_p1[i0] * s_p2[i1];
        }
    }
}

extern "C" void kernel_launch(void* const* d_in, const int* in_sizes, int n_in,
                              void* d_out, int out_size, void* d_ws, size_t ws_size,
                              hipStream_t stream) {
    // setup_inputs order:
    // 0: bin_centers (1024 f32)  1: lb1  2: ub1  3: lb2  4: ub2  5: resp (unused)
    // 6: obs_idx (N*2 int32)     7: operator_number (unused)
    const float* bin_centers = (const float*)d_in[0];
    const float* lb1 = (const float*)d_in[1];
    const float* ub1 = (const float*)d_in[2];
    const float* lb2 = (const float*)d_in[3];
    const float* ub2 = (const float*)d_in[4];
    const int*   obs_idx = (const int*)d_in[6];
    float* out = (float*)d_out;

    const int n   = out_size;                 // N_DIFFS
    const int n4  = n >> 2;
    const int spb = TPB * (OPT / 4);          // 2048 v4f slots per block
    int grid = (n4 + spb - 1) / spb;
    if (grid < 1) grid = 1;

    PerformanceModel_kernel<<<grid, TPB, 0, stream>>>(
        bin_centers, lb1, ub1, lb2, ub2, obs_idx, out, n);
}